// RoiPoolingConv_19086834663495
// MI455X (gfx1250) — compile-verified
//
#include <hip/hip_runtime.h>
#include <cstdint>

// Problem constants (match reference module).
#define IMG_H 256
#define IMG_W 256
#define IMG_C 512
#define POOLP 7
#define NCELL (POOLP * POOLP)   // 49
#define NTHR  (IMG_C / 4)       // 128 lanes, one float4 of channels each

// Builtin signature (clang gfx1250): void(v4i AS1*, v4i AS3*, Ii offset, Ii cpol)
typedef int v4i __attribute__((ext_vector_type(4)));
typedef __attribute__((address_space(1))) v4i v4i_g;  // global
typedef __attribute__((address_space(3))) v4i v4i_l;  // LDS

// Async copy of 16 bytes global -> LDS (tracked by ASYNCcnt).
__device__ __forceinline__ void async_copy_b128(const float* gsrc, float4* ldst) {
#if __has_builtin(__builtin_amdgcn_global_load_async_to_lds_b128)
    __builtin_amdgcn_global_load_async_to_lds_b128((v4i_g*)gsrc, (v4i_l*)ldst,
                                                   /*offset=*/0, /*cpol=*/0);
#else
    unsigned lofs = (unsigned)(uintptr_t)ldst;            // low 32 bits = LDS byte address
    unsigned long long g = (unsigned long long)(uintptr_t)gsrc;
    asm volatile("global_load_async_to_lds_b128 %0, %1, off"
                 :: "v"(lofs), "v"(g) : "memory");
#endif
}

__device__ __forceinline__ void wait_async0() {
#if __has_builtin(__builtin_amdgcn_s_wait_asynccnt)
    __builtin_amdgcn_s_wait_asynccnt(0);
    asm volatile("" ::: "memory");   // compiler ordering fence for the LDS reads
#else
    asm volatile("s_wait_asynccnt 0x0" ::: "memory");
#endif
}

__global__ void __launch_bounds__(NTHR)
roi_pool_bilinear_kernel(const float* __restrict__ img,
                         const int*   __restrict__ rois,
                         float*       __restrict__ out) {
    __shared__ float4 tile[4][NTHR];   // 4 corner rows x 512 channels = 8 KB

    const int cell = blockIdx.x;       // 0..48  -> (py, px)
    const int r    = blockIdx.y;       // roi index
    const int py   = cell / POOLP;
    const int px   = cell % POOLP;
    const int t    = threadIdx.x;      // 0..127 channel group

    // ROI (x, y, w, h) — uniform per block (scalar loads).
    const int rx = rois[4 * r + 0];
    const int ry = rois[4 * r + 1];
    const int rw = rois[4 * r + 2];
    const int rh = rois[4 * r + 3];
    const float wf = (float)rw, hf = (float)rh;

    // Half-pixel-center source coords, clipped (matches TF2 bilinear resize).
    float sy = ((float)py + 0.5f) * hf / (float)POOLP - 0.5f;
    sy = fminf(fmaxf(sy, 0.0f), hf - 1.0f);
    int   y0 = (int)floorf(sy);
    int   y1 = min(y0 + 1, rh - 1);
    float wy = sy - (float)y0;

    float sx = ((float)px + 0.5f) * wf / (float)POOLP - 0.5f;
    sx = fminf(fmaxf(sx, 0.0f), wf - 1.0f);
    int   x0 = (int)floorf(sx);
    int   x1 = min(x0 + 1, rw - 1);
    float wx = sx - (float)x0;

    // Absolute corner-pixel channel-vector base offsets.
    const long long row0 = (long long)(y0 + ry) * IMG_W;
    const long long row1 = (long long)(y1 + ry) * IMG_W;
    const long long c00  = (row0 + (x0 + rx)) * IMG_C;
    const long long c01  = (row0 + (x1 + rx)) * IMG_C;
    const long long c10  = (row1 + (x0 + rx)) * IMG_C;
    const long long c11  = (row1 + (x1 + rx)) * IMG_C;

    // Stage the 4 corner rows into LDS with the async data mover.
    // Each lane moves its own float4 of channels; a wave's 32 lanes form one
    // fully-coalesced 512B transaction; all 4 rows are in flight concurrently.
    const int cb = t * 4;
    async_copy_b128(img + c00 + cb, &tile[0][t]);
    async_copy_b128(img + c01 + cb, &tile[1][t]);
    async_copy_b128(img + c10 + cb, &tile[2][t]);
    async_copy_b128(img + c11 + cb, &tile[3][t]);
    wait_async0();   // per-wave ASYNCcnt: each lane reads exactly what it loaded

    const float4 p00 = tile[0][t];
    const float4 p01 = tile[1][t];
    const float4 p10 = tile[2][t];
    const float4 p11 = tile[3][t];

    const float owx = 1.0f - wx;
    const float owy = 1.0f - wy;

    float4 res;
    {
        float tx, bx;
        tx = p00.x * owx + p01.x * wx;  bx = p10.x * owx + p11.x * wx;
        res.x = tx * owy + bx * wy;
        tx = p00.y * owx + p01.y * wx;  bx = p10.y * owx + p11.y * wx;
        res.y = tx * owy + bx * wy;
        tx = p00.z * owx + p01.z * wx;  bx = p10.z * owx + p11.z * wx;
        res.z = tx * owy + bx * wy;
        tx = p00.w * owx + p01.w * wx;  bx = p10.w * owx + p11.w * wx;
        res.w = tx * owy + bx * wy;
    }

    // out[(r*49 + cell)*512 + 4t .. +3]
    float4* out4 = (float4*)out;
    out4[((long long)r * NCELL + cell) * NTHR + t] = res;
}

extern "C" void kernel_launch(void* const* d_in, const int* in_sizes, int n_in,
                              void* d_out, int out_size, void* d_ws, size_t ws_size,
                              hipStream_t stream) {
    const float* img  = (const float*)d_in[0];
    const int*   rois = (const int*)d_in[1];
    // d_in[2] is pool_size (device scalar == 7); POOLP is compile-time.
    const int R = in_sizes[1] / 4;      // 300 ROIs

    dim3 grid(NCELL, R);
    roi_pool_bilinear_kernel<<<grid, NTHR, 0, stream>>>(img, rois, (float*)d_out);
}